// SpectralAttention_10402410791621
// MI455X (gfx1250) — compile-verified
//
#include <hip/hip_runtime.h>

// Problem constants (match reference)
#define B_   4
#define T_   2048
#define D_   1024
#define K_   64
#define CH   64          // chunk length
#define NC   (T_/CH)     // 32 chunks
#define QKVN 192         // Q(64) | Kf(64) | V(64)
#define MROWS (B_*T_)    // 8192

typedef float v2f __attribute__((ext_vector_type(2)));
typedef float v8f __attribute__((ext_vector_type(8)));

// D = A(16x4 f32) x B(4x16 f32) + C(16x16 f32), full fp32 precision
#define WMMA_F32(a,b,c) \
  __builtin_amdgcn_wmma_f32_16x16x4_f32(false,(a),false,(b),(short)0,(c),false,false)

// ---------------------------------------------------------------------------
// Kernel 0: fold spectral filters into the pre-projection.
//   Wqkv[d][0:64)   = sum_t W_pre[d][t] * Q_filt[t][j]
//   Wqkv[d][64:128) = sum_t W_pre[d][t] * K_filt[t][j]
//   Wqkv[d][128:192)= W_v[d][j]
//   bqkv analogous from b_pre / b_v.
// ---------------------------------------------------------------------------
__global__ void __launch_bounds__(256)
fuse_weights_kernel(const float* __restrict__ W_pre, const float* __restrict__ b_pre,
                    const float* __restrict__ Qf, const float* __restrict__ Kf,
                    const float* __restrict__ Wv, const float* __restrict__ bv,
                    float* __restrict__ Wqkv, float* __restrict__ bqkv) {
  const int NW = D_ * QKVN;
  int idx = blockIdx.x * 256 + threadIdx.x;
  if (idx < NW) {
    int d = idx / QKVN, j = idx % QKVN;
    float acc = 0.f;
    if (j < 64) {
      const float* w = W_pre + d * T_;
      const float* f = Qf + j;
      for (int t = 0; t < T_; ++t) acc += w[t] * f[t * K_];
    } else if (j < 128) {
      const float* w = W_pre + d * T_;
      const float* f = Kf + (j - 64);
      for (int t = 0; t < T_; ++t) acc += w[t] * f[t * K_];
    } else {
      acc = Wv[d * K_ + (j - 128)];
    }
    Wqkv[idx] = acc;
  } else if (idx < NW + QKVN) {
    int j = idx - NW;
    float acc = 0.f;
    if (j < 64)       { for (int t = 0; t < T_; ++t) acc += b_pre[t] * Qf[t * K_ + j]; }
    else if (j < 128) { for (int t = 0; t < T_; ++t) acc += b_pre[t] * Kf[t * K_ + (j - 64)]; }
    else              { acc = bv[j - 128]; }
    bqkv[j] = acc;
  }
}

// ---------------------------------------------------------------------------
// Kernel 1: QKV = x[8192,1024] @ Wqkv[1024,192] + bqkv.  WMMA f32 16x16x4.
// Grid (512, 3), block 128 (4 waves); each wave owns one 16x16 output tile.
// ---------------------------------------------------------------------------
__global__ void __launch_bounds__(128)
qkv_gemm_kernel(const float* __restrict__ X, const float* __restrict__ Wqkv,
                const float* __restrict__ bqkv, float* __restrict__ QKV) {
  const int lane = threadIdx.x & 31;
  const int wave = threadIdx.x >> 5;
  const int m_l  = lane & 15;                 // M (or N) index inside tile
  const int kp   = (lane >> 4) << 1;          // reduction pair: 0 or 2
  const int row0 = blockIdx.x * 16;
  const int col0 = (blockIdx.y * 4 + wave) * 16;

  const float* arow = X + (size_t)(row0 + m_l) * D_;
  const float* bcol = Wqkv + col0 + m_l;
  v8f c = {};
  for (int kk = 0; kk < D_; kk += 4) {
    v2f a, b;
    a.x = arow[kk + kp];              a.y = arow[kk + kp + 1];
    b.x = bcol[(kk + kp) * QKVN];     b.y = bcol[(kk + kp + 1) * QKVN];
    c = WMMA_F32(a, b, c);
  }
  const float bias = bqkv[col0 + m_l];
  const int mbase = row0 + ((lane >> 4) << 3);
  float* outp = QKV + col0 + m_l;
  for (int r = 0; r < 8; ++r)
    outp[(size_t)(mbase + r) * QKVN] = c[r] + bias;
}

// ---------------------------------------------------------------------------
// Kernel 2a: per-chunk state  S[b,c] = (decay*V)^T @ Kf   (64x64, reduce s=64)
// Grid 128 (=B*NC), block 256 (8 waves, 2 tiles each).
// ---------------------------------------------------------------------------
__global__ void __launch_bounds__(256)
chunk_state_kernel(const float* __restrict__ QKV, const float* __restrict__ decay,
                   float* __restrict__ S) {
  const int bc   = blockIdx.x;                 // b*NC + c
  const int rowb = bc * CH;                    // global row (b*T + c*CH)
  const int crow = (bc & (NC - 1)) * CH;       // time index within batch
  const int lane = threadIdx.x & 31;
  const int wave = threadIdx.x >> 5;
  const int m_l  = lane & 15;
  const int kp   = (lane >> 4) << 1;

  for (int q = 0; q < 2; ++q) {
    const int tile = wave * 2 + q;
    const int row0 = (tile >> 2) * 16;         // k
    const int col0 = (tile & 3) * 16;          // n
    v8f c = {};
    for (int kk = 0; kk < CH; kk += 4) {
      const int s0 = kk + kp;
      v2f a, b;
      a.x = decay[crow + s0]     * QKV[(size_t)(rowb + s0)     * QKVN + 128 + row0 + m_l];
      a.y = decay[crow + s0 + 1] * QKV[(size_t)(rowb + s0 + 1) * QKVN + 128 + row0 + m_l];
      b.x = QKV[(size_t)(rowb + s0)     * QKVN + 64 + col0 + m_l];
      b.y = QKV[(size_t)(rowb + s0 + 1) * QKVN + 64 + col0 + m_l];
      c = WMMA_F32(a, b, c);
    }
    const int mbase = row0 + ((lane >> 4) << 3);
    float* out = S + (size_t)bc * (K_ * K_) + col0 + m_l;
    for (int r = 0; r < 8; ++r) out[(mbase + r) * K_] = c[r];
  }
}

// ---------------------------------------------------------------------------
// Kernel 2b: exclusive prefix over chunks: Hpre[b,c] = sum_{c'<c} S[b,c'].
// One thread per (b, k*n) element; coalesced per chunk step.
// ---------------------------------------------------------------------------
__global__ void __launch_bounds__(256)
chunk_prefix_kernel(const float* __restrict__ S, float* __restrict__ Hpre) {
  int idx = blockIdx.x * 256 + threadIdx.x;    // B * 4096
  int b  = idx >> 12;
  int kn = idx & 4095;
  float acc = 0.f;
  for (int cc = 0; cc < NC; ++cc) {
    size_t o = (size_t)(b * NC + cc) * (K_ * K_) + kn;
    Hpre[o] = acc;
    acc += S[o];
  }
}

// ---------------------------------------------------------------------------
// Kernel 2c: per-chunk output
//   P = Q @ V^T  (64x64),  Pm = causal-mask(P) * decay[s]   -> LDS
//   Y = Pm @ Kf + Q @ Hpre
// Grid 128, block 256; Pm staged via LDS between the two WMMA phases.
// ---------------------------------------------------------------------------
__global__ void __launch_bounds__(256)
chunk_output_kernel(const float* __restrict__ QKV, const float* __restrict__ decay,
                    const float* __restrict__ Hpre, float* __restrict__ Y) {
  __shared__ float Pm[64 * 72];                // padded rows (72) vs bank conflicts
  const int bc   = blockIdx.x;
  const int rowb = bc * CH;
  const int crow = (bc & (NC - 1)) * CH;
  const int lane = threadIdx.x & 31;
  const int wave = threadIdx.x >> 5;
  const int m_l  = lane & 15;
  const int kp   = (lane >> 4) << 1;

  // ---- phase 1: P = Q @ V^T, mask + decay, to LDS ----
  for (int q = 0; q < 2; ++q) {
    const int tile = wave * 2 + q;
    const int row0 = (tile >> 2) * 16;         // t
    const int col0 = (tile & 3) * 16;          // s
    v8f c = {};
    const float* qrow = QKV + (size_t)(rowb + row0 + m_l) * QKVN;        // Q[t][*]
    const float* vrow = QKV + (size_t)(rowb + col0 + m_l) * QKVN + 128;  // V[s][*]
    for (int kk = 0; kk < K_; kk += 4) {
      v2f a, b;
      a.x = qrow[kk + kp]; a.y = qrow[kk + kp + 1];
      b.x = vrow[kk + kp]; b.y = vrow[kk + kp + 1];
      c = WMMA_F32(a, b, c);
    }
    const int mbase = row0 + ((lane >> 4) << 3);
    for (int r = 0; r < 8; ++r) {
      int t = mbase + r, s = col0 + m_l;
      Pm[t * 72 + s] = (s <= t) ? c[r] * decay[crow + s] : 0.f;
    }
  }
  __syncthreads();

  // ---- phase 2: Y = Pm @ Kf + Q @ Hpre ----
  for (int q = 0; q < 2; ++q) {
    const int tile = wave * 2 + q;
    const int row0 = (tile >> 2) * 16;         // t
    const int col0 = (tile & 3) * 16;          // n
    v8f c = {};
    const float* prow = Pm + (row0 + m_l) * 72;
    const float* kcol = QKV + 64 + col0 + m_l; // Kf column, row = rowb + s
    for (int kk = 0; kk < CH; kk += 4) {
      v2f a, b;
      a.x = prow[kk + kp];                    a.y = prow[kk + kp + 1];
      b.x = kcol[(size_t)(rowb + kk + kp) * QKVN];
      b.y = kcol[(size_t)(rowb + kk + kp + 1) * QKVN];
      c = WMMA_F32(a, b, c);
    }
    const float* qrow = QKV + (size_t)(rowb + row0 + m_l) * QKVN;
    const float* hcol = Hpre + (size_t)bc * (K_ * K_) + col0 + m_l;
    for (int kk = 0; kk < K_; kk += 4) {
      v2f a, b;
      a.x = qrow[kk + kp];        a.y = qrow[kk + kp + 1];
      b.x = hcol[(kk + kp) * K_]; b.y = hcol[(kk + kp + 1) * K_];
      c = WMMA_F32(a, b, c);
    }
    const int mbase = row0 + ((lane >> 4) << 3);
    float* out = Y + (size_t)rowb * K_ + col0 + m_l;
    for (int r = 0; r < 8; ++r) out[(mbase + r) * K_] = c[r];
  }
}

// ---------------------------------------------------------------------------
// Kernel 3: Out = Y[8192,64] @ W_o[64,1024] + b_o.
// Grid (512, 8), block 256 (8 waves, one 16x16 tile each).
// ---------------------------------------------------------------------------
__global__ void __launch_bounds__(256)
out_gemm_kernel(const float* __restrict__ Y, const float* __restrict__ Wo,
                const float* __restrict__ bo, float* __restrict__ Out) {
  const int lane = threadIdx.x & 31;
  const int wave = threadIdx.x >> 5;
  const int m_l  = lane & 15;
  const int kp   = (lane >> 4) << 1;
  const int row0 = blockIdx.x * 16;
  const int col0 = blockIdx.y * 128 + wave * 16;

  const float* yrow = Y + (size_t)(row0 + m_l) * K_;
  const float* wcol = Wo + col0 + m_l;
  v8f c = {};
  for (int kk = 0; kk < K_; kk += 4) {
    v2f a, b;
    a.x = yrow[kk + kp];          a.y = yrow[kk + kp + 1];
    b.x = wcol[(kk + kp) * D_];   b.y = wcol[(kk + kp + 1) * D_];
    c = WMMA_F32(a, b, c);
  }
  const float bias = bo[col0 + m_l];
  const int mbase = row0 + ((lane >> 4) << 3);
  float* out = Out + col0 + m_l;
  for (int r = 0; r < 8; ++r)
    out[(size_t)(mbase + r) * D_] = c[r] + bias;
}

// ---------------------------------------------------------------------------
extern "C" void kernel_launch(void* const* d_in, const int* in_sizes, int n_in,
                              void* d_out, int out_size, void* d_ws, size_t ws_size,
                              hipStream_t stream) {
  const float* x     = (const float*)d_in[0];
  const float* W_pre = (const float*)d_in[1];
  const float* b_pre = (const float*)d_in[2];
  const float* Qf    = (const float*)d_in[3];
  const float* Kf    = (const float*)d_in[4];
  const float* Wv    = (const float*)d_in[5];
  const float* bv    = (const float*)d_in[6];
  const float* Wo    = (const float*)d_in[7];
  const float* bo    = (const float*)d_in[8];
  const float* decay = (const float*)d_in[9];
  float* out = (float*)d_out;

  // Workspace layout (floats), ~12.8 MB total
  float* ws    = (float*)d_ws;
  float* Wqkv  = ws;                       // 196608
  float* bqkv  = ws + 196608;              // 192 (pad to 196864)
  float* QKV   = ws + 196864;              // 8192*192 = 1572864
  float* S     = ws + 196864 + 1572864;    // 4*32*4096 = 524288
  float* Hpre  = S + 524288;               // 524288
  float* Yb    = Hpre + 524288;            // 8192*64 = 524288

  (void)in_sizes; (void)n_in; (void)out_size; (void)ws_size;

  // 0) fold filters into fused QKV projection weights
  fuse_weights_kernel<<<(D_ * QKVN + QKVN + 255) / 256, 256, 0, stream>>>(
      W_pre, b_pre, Qf, Kf, Wv, bv, Wqkv, bqkv);

  // 1) QKV projection (WMMA f32)
  qkv_gemm_kernel<<<dim3(MROWS / 16, 3), 128, 0, stream>>>(x, Wqkv, bqkv, QKV);

  // 2) chunked causal linear attention
  chunk_state_kernel<<<B_ * NC, 256, 0, stream>>>(QKV, decay, S);
  chunk_prefix_kernel<<<(B_ * K_ * K_) / 256, 256, 0, stream>>>(S, Hpre);
  chunk_output_kernel<<<B_ * NC, 256, 0, stream>>>(QKV, decay, Hpre, Yb);

  // 3) output projection
  out_gemm_kernel<<<dim3(MROWS / 16, D_ / 128), 256, 0, stream>>>(Yb, Wo, bo, out);
}